// CompositeLoss_28089086116639
// MI455X (gfx1250) — compile-verified
//
#include <hip/hip_runtime.h>
#include <hip/hip_bf16.h>
#include <math.h>

// ---------------- problem constants (from reference) ----------------
#define BB 2
#define SS 1024
#define VV 32000
#define NN 100
#define MM 25
#define ROWS (BB * SS)          // 2048
#define FULL_CHUNKS 31          // 31 * 256 float4 = 7936; tail = 64 float4
#define TAIL_BASE 7936
#define DEPTH 4                 // async pipeline depth (ring of 4 LDS buffers)

#define CLS_W   0.0f
#define COORD_W 0.7f
#define IOU_W   0.75f
#define L1_W    0.25f
#define LM_W    0.2f
#define DET_W   0.8f
#define EPSF    1e-7f
#define NEGF    (-1e9f)

// ---------------- CDNA5 async global->LDS path (guarded) ----------------
#if defined(__HIP_DEVICE_COMPILE__) && defined(__gfx1250__) && \
    __has_builtin(__builtin_amdgcn_global_load_async_to_lds_b128)
#define USE_ASYNC 1
#else
#define USE_ASYNC 0
#endif

#if USE_ASYNC
typedef int v4i __attribute__((vector_size(16)));
typedef __attribute__((address_space(1))) v4i gas_v4i;   // global
typedef __attribute__((address_space(3))) v4i las_v4i;   // LDS

__device__ __forceinline__ void async_copy_b128(const void* g, void* l) {
  __builtin_amdgcn_global_load_async_to_lds_b128(
      (gas_v4i*)(void*)g,
      (las_v4i*)l,
      0, 0);
}
#if __has_builtin(__builtin_amdgcn_s_wait_asynccnt)
#define WAIT_ASYNC(n) do { __builtin_amdgcn_s_wait_asynccnt(n); asm volatile("" ::: "memory"); } while (0)
#else
#define WAIT_ASYNC(n) asm volatile("s_wait_asynccnt %0" :: "i"(n) : "memory")
#endif
#endif

// branchless online softmax update: exactly one v_exp_f32, no divergence.
// rescale factor is e = exp(-|x-m|) on one side of the max and 1.0 on the other:
//   x >  m : s = s*e + 1 ; m = x
//   x <= m : s = s*1 + e ; m unchanged
__device__ __forceinline__ void osm_update(float& m, float& s, float x) {
  float e  = __expf(-fabsf(x - m));
  bool big = x > m;
  float a  = big ? e : 1.0f;
  float b  = big ? 1.0f : e;
  s = fmaf(s, a, b);
  m = fmaxf(m, x);
}

__device__ __forceinline__ void osm_update4(float& m, float& s, float4 x) {
  osm_update(m, s, x.x); osm_update(m, s, x.y);
  osm_update(m, s, x.z); osm_update(m, s, x.w);
}

// ---------------- kernel 1: per-row online log-sum-exp ----------------
__global__ __launch_bounds__(256) void lm_row_kernel(const float* __restrict__ logits,
                                                     const int* __restrict__ labels,
                                                     float* __restrict__ row_loss,
                                                     float* __restrict__ row_msk) {
  const int row = blockIdx.x;
  const int tid = threadIdx.x;
  const float* rp = logits + (size_t)row * VV;

  float m = -INFINITY, s = 0.0f;

#if USE_ASYNC
  __shared__ float4 stage[DEPTH][256];   // 16 KB ring, one 16B slot per lane per buffer
  // prologue: fill the ring (chunks 0..3)
#pragma unroll
  for (int j = 0; j < DEPTH; ++j)
    async_copy_b128(rp + ((size_t)j * 256 + tid) * 4, &stage[j][tid]);

  // steady state: consume chunk j, then refill slot j&3 with chunk j+4
  for (int j = 0; j < FULL_CHUNKS - DEPTH; ++j) {      // j = 0..26
    WAIT_ASYNC(DEPTH - 1);           // completed through chunk j (in-order completion)
    float4 x = stage[j & (DEPTH - 1)][tid];
    osm_update4(m, s, x);
    asm volatile("" ::: "memory");   // keep refill after the consume read
    async_copy_b128(rp + ((size_t)(j + DEPTH) * 256 + tid) * 4,
                    &stage[j & (DEPTH - 1)][tid]);
  }
  // drain: chunks 27..30 live in slots 3,0,1,2; waits 3,2,1,0
  { WAIT_ASYNC(3); float4 x = stage[3][tid]; osm_update4(m, s, x); }
  { WAIT_ASYNC(2); float4 x = stage[0][tid]; osm_update4(m, s, x); }
  { WAIT_ASYNC(1); float4 x = stage[1][tid]; osm_update4(m, s, x); }
  { WAIT_ASYNC(0); float4 x = stage[2][tid]; osm_update4(m, s, x); }
#else
  for (int j = 0; j < FULL_CHUNKS; ++j) {
    float4 x = *(const float4*)(rp + ((size_t)j * 256 + tid) * 4);
    osm_update4(m, s, x);
  }
#endif
  // tail: last 64 float4 (256 floats), direct loads
  if (tid < 64) {
    float4 x = *(const float4*)(rp + ((size_t)TAIL_BASE + tid) * 4);
    osm_update4(m, s, x);
  }

  // wave32 combine of (m, s)
  for (int off = 16; off > 0; off >>= 1) {
    float om = __shfl_xor(m, off, 32);
    float os = __shfl_xor(s, off, 32);
    float nm = fmaxf(m, om);
    s = s * expf(m - nm) + os * expf(om - nm);
    m = nm;
  }
  __shared__ float wm[8], wsm[8];
  if ((tid & 31) == 0) { wm[tid >> 5] = m; wsm[tid >> 5] = s; }
  __syncthreads();
  if (tid == 0) {
    float M0 = wm[0], S0 = wsm[0];
    for (int w = 1; w < 8; ++w) {
      float nm = fmaxf(M0, wm[w]);
      S0 = S0 * expf(M0 - nm) + wsm[w] * expf(wm[w] - nm);
      M0 = nm;
    }
    int lab = labels[row];
    float loss = 0.0f, msk = 0.0f;
    if (lab != -100) {
      int lc = min(max(lab, 0), VV - 1);
      loss = M0 + logf(S0) - rp[lc];
      msk = 1.0f;
    }
    row_loss[row] = loss;
    row_msk[row]  = msk;
  }
}

// ---------------- kernel 2: deterministic reduction over rows ----------------
__global__ __launch_bounds__(256) void lm_reduce_kernel(const float* __restrict__ row_loss,
                                                        const float* __restrict__ row_msk,
                                                        float* __restrict__ out_lm) {
  const int tid = threadIdx.x;
  float a = 0.0f, c = 0.0f;
  for (int j = 0; j < ROWS / 256; ++j) {
    int i = tid * (ROWS / 256) + j;     // fixed mapping -> deterministic
    a += row_loss[i];
    c += row_msk[i];
  }
  for (int off = 16; off > 0; off >>= 1) {
    a += __shfl_xor(a, off, 32);
    c += __shfl_xor(c, off, 32);
  }
  __shared__ float sa[8], sc[8];
  if ((tid & 31) == 0) { sa[tid >> 5] = a; sc[tid >> 5] = c; }
  __syncthreads();
  if (tid == 0) {
    float A = 0.0f, C = 0.0f;
    for (int w = 0; w < 8; ++w) { A += sa[w]; C += sc[w]; }
    out_lm[0] = A / fmaxf(C, 1.0f);
  }
}

// ---------------- kernel 3: detection loss, one block per image ----------------
__global__ __launch_bounds__(256) void det_kernel(const float* __restrict__ box_preds,
                                                  const int* __restrict__ tgt_labels,
                                                  const float* __restrict__ tgt_boxes,
                                                  float* __restrict__ det_out) {
  const int b   = blockIdx.x;
  const int tid = threadIdx.x;

  __shared__ float pc[NN][4];
  __shared__ float tc[MM][4];
  __shared__ float validf[MM];
  __shared__ float ious[NN * MM];
  __shared__ float rv[256];
  __shared__ int   ri[256];
  __shared__ int   mps[MM], mts[MM];
  __shared__ float mok[MM];

  if (tid < NN) {
    const float* bp = box_preds + ((size_t)b * NN + tid) * 4;
    float x = bp[0], y = bp[1], w = bp[2], h = bp[3];
    pc[tid][0] = x; pc[tid][1] = y; pc[tid][2] = x + w; pc[tid][3] = y + h;
  }
  if (tid < MM) {
    const float* tb = tgt_boxes + ((size_t)b * MM + tid) * 4;
    float x = tb[0], y = tb[1], w = tb[2], h = tb[3];
    tc[tid][0] = x; tc[tid][1] = y; tc[tid][2] = x + w; tc[tid][3] = y + h;
    int lab = tgt_labels[b * MM + tid];
    validf[tid] = (lab != -100 && w > 0.0f && h > 0.0f) ? 1.0f : 0.0f;
  }
  __syncthreads();

  // pairwise IoU, masked by target validity
  for (int idx = tid; idx < NN * MM; idx += 256) {
    int p = idx / MM, t = idx % MM;
    float ltx = fmaxf(pc[p][0], tc[t][0]), lty = fmaxf(pc[p][1], tc[t][1]);
    float rbx = fminf(pc[p][2], tc[t][2]), rby = fminf(pc[p][3], tc[t][3]);
    float iw = fmaxf(rbx - ltx, 0.0f), ih = fmaxf(rby - lty, 0.0f);
    float inter = iw * ih;
    float aa = (pc[p][2] - pc[p][0]) * (pc[p][3] - pc[p][1]);
    float ab = (tc[t][2] - tc[t][0]) * (tc[t][3] - tc[t][1]);
    float uni = aa + ab - inter;
    float iou = inter / fmaxf(uni, EPSF);
    ious[idx] = (validf[t] > 0.0f) ? iou : NEGF;
  }
  __syncthreads();

  // greedy matching, K = min(N, M) = 25; replicate jnp.argmax first-flat-index tie-break
  for (int k = 0; k < MM; ++k) {
    float best = -INFINITY; int bidx = 0;
    for (int idx = tid; idx < NN * MM; idx += 256) {
      float v = ious[idx];
      if (v > best) { best = v; bidx = idx; }   // keeps earliest idx per thread
    }
    rv[tid] = best; ri[tid] = bidx;
    __syncthreads();
    for (int str = 128; str > 0; str >>= 1) {
      if (tid < str) {
        float v2 = rv[tid + str]; int i2 = ri[tid + str];
        if (v2 > rv[tid] || (v2 == rv[tid] && i2 < ri[tid])) { rv[tid] = v2; ri[tid] = i2; }
      }
      __syncthreads();
    }
    if (tid == 0) {
      int idx = ri[0]; float val = rv[0];
      mps[k] = idx / MM; mts[k] = idx % MM;
      mok[k] = (val >= 0.5f) ? 1.0f : 0.0f;
    }
    __syncthreads();
    int p = mps[k], t = mts[k];
    if (tid < MM) ious[p * MM + tid] = NEGF;
    if (tid < NN) ious[tid * MM + t] = NEGF;
    __syncthreads();
  }

  if (tid == 0) {
    float matched = 0.0f, nmatch = 0.0f, nvalid = 0.0f;
    for (int t = 0; t < MM; ++t) nvalid += validf[t];
    for (int k = 0; k < MM; ++k) {
      int p = mps[k], t = mts[k];
      float ok = mok[k];
      float ax1 = pc[p][0], ay1 = pc[p][1], ax2 = pc[p][2], ay2 = pc[p][3];
      float bx1 = tc[t][0], by1 = tc[t][1], bx2 = tc[t][2], by2 = tc[t][3];
      // GIoU loss on corner pairs
      float ltx = fmaxf(ax1, bx1), lty = fmaxf(ay1, by1);
      float rbx = fminf(ax2, bx2), rby = fminf(ay2, by2);
      float iw = fmaxf(rbx - ltx, 0.0f), ih = fmaxf(rby - lty, 0.0f);
      float inter = iw * ih;
      float aa = (ax2 - ax1) * (ay2 - ay1);
      float ab = (bx2 - bx1) * (by2 - by1);
      float uni = aa + ab - inter;
      float iou = inter / (uni + EPSF);
      float cw = fmaxf(fmaxf(ax2, bx2) - fminf(ax1, bx1), 0.0f);
      float ch = fmaxf(fmaxf(ay2, by2) - fminf(ay1, by1), 0.0f);
      float ac = cw * ch;
      float gl = 1.0f - (iou - (ac - uni) / (ac + EPSF));
      // smooth L1 over the 4 corner coords (beta = 1), mean
      float l1 = 0.0f;
      float dv[4] = { fabsf(ax1 - bx1), fabsf(ay1 - by1), fabsf(ax2 - bx2), fabsf(ay2 - by2) };
      for (int c = 0; c < 4; ++c) {
        float d = dv[c];
        l1 += (d < 1.0f) ? 0.5f * d * d : (d - 0.5f);
      }
      l1 *= 0.25f;
      matched += ok * (COORD_W * (IOU_W * gl + L1_W * l1));   // CLS_W == 0 -> CE term drops
      nmatch  += ok;
    }
    float pen = 0.2f * COORD_W * L1_W + 0.2f * CLS_W;         // 0.035
    float unmatched = ((float)NN - nmatch) + (nvalid - nmatch);
    det_out[b] = matched + pen * unmatched;
  }
}

// ---------------- kernel 4: combine ----------------
__global__ void finalize_kernel(const float* __restrict__ lm,
                                const float* __restrict__ det,
                                float* __restrict__ out) {
  out[0] = LM_W * lm[0] + DET_W * 0.5f * (det[0] + det[1]);
}

extern "C" void kernel_launch(void* const* d_in, const int* in_sizes, int n_in,
                              void* d_out, int out_size, void* d_ws, size_t ws_size,
                              hipStream_t stream) {
  const float* lm_logits   = (const float*)d_in[0];
  const int*   lm_labels   = (const int*)d_in[1];
  // d_in[2] = class_logits: unused (CLS_W == 0)
  const float* box_preds   = (const float*)d_in[3];
  const int*   tgt_labels  = (const int*)d_in[4];
  const float* tgt_boxes   = (const float*)d_in[5];

  float* ws       = (float*)d_ws;
  float* row_loss = ws;                 // [2048]
  float* row_msk  = ws + ROWS;          // [2048]
  float* lm_scal  = ws + 2 * ROWS;      // [1]
  float* det_scal = ws + 2 * ROWS + 1;  // [2]

  lm_row_kernel<<<ROWS, 256, 0, stream>>>(lm_logits, lm_labels, row_loss, row_msk);
  lm_reduce_kernel<<<1, 256, 0, stream>>>(row_loss, row_msk, lm_scal);
  det_kernel<<<BB, 256, 0, stream>>>(box_preds, tgt_labels, tgt_boxes, det_scal);
  finalize_kernel<<<1, 1, 0, stream>>>(lm_scal, det_scal, (float*)d_out);
}